// YOLOv1Loss_20684562497969
// MI455X (gfx1250) — compile-verified
//
#include <hip/hip_runtime.h>
#include <math.h>

// ---- problem constants (from reference) ----
#define NBATCH 16
#define NANCH  3
#define NGRID  52
#define NCELLS (NGRID * NGRID)        // 2704
#define NPRED  (NANCH * NCELLS)       // 8112
#define NCH    15                     // 5 * NA
#define GTB    16                     // GTs per batch
#define NEGV   (-1e30f)
#define EPSV   (1e-16f)
#define NBLK3  508                    // 507 blocks cover 16*8112 preds exactly; +1 zero pad row (multiple of 4)

typedef __attribute__((ext_vector_type(2))) float v2f;
typedef __attribute__((ext_vector_type(8))) float v8f;

__device__ __forceinline__ float sigf(float x) { return 1.0f / (1.0f + expf(-x)); }

// ------------------------------------------------------------------
// Kernel 1: decode predictions + pairwise IoU / GIoU vs the 16 GTs of
// the block's batch. grid = NBATCH * 32 blocks of 256.
// ------------------------------------------------------------------
__global__ void __launch_bounds__(256)
k_decode(const float* __restrict__ outp, const float* __restrict__ gts,
         float* __restrict__ gous, float* __restrict__ ious,
         float* __restrict__ boxesW, float* __restrict__ confW)
{
    const int b   = blockIdx.x >> 5;      // / 32
    const int blk = blockIdx.x & 31;
    const int t   = threadIdx.x;

    __shared__ float gx1[GTB], gy1[GTB], gx2[GTB], gy2[GTB], gar[GTB];
    if (t < GTB) {
        const float* g = gts + (size_t)(b * GTB + t) * 5;
        float cx = g[1], cy = g[2], w = g[3], h = g[4];
        float x1 = cx - w * 0.5f, x2 = cx + w * 0.5f;
        float y1 = cy - h * 0.5f, y2 = cy + h * 0.5f;
        gx1[t] = x1; gx2[t] = x2; gy1[t] = y1; gy2[t] = y2;
        gar[t] = (x2 - x1) * (y2 - y1);
    }
    __syncthreads();

    const int i = blk * 256 + t;
    if (i >= NPRED) return;

    const int a   = i / NCELLS;
    const int rem = i - a * NCELLS;
    const int gy  = rem / NGRID;
    const int gx  = rem - gy * NGRID;
    const int cell = gy * NGRID + gx;

    const float* base = outp + (size_t)b * NCH * NCELLS;
    const float t0 = base[(size_t)(a * 5 + 0) * NCELLS + cell];
    const float t1 = base[(size_t)(a * 5 + 1) * NCELLS + cell];
    const float t2 = base[(size_t)(a * 5 + 2) * NCELLS + cell];
    const float t3 = base[(size_t)(a * 5 + 3) * NCELLS + cell];
    const float t4 = base[(size_t)(a * 5 + 4) * NCELLS + cell];

    const float xs = (sigf(t0) + (float)gx) / (float)NGRID;   // / nh (nh == 52)
    const float ys = (sigf(t1) + (float)gy) / (float)NGRID;
    const float wz = expf(-t2 * t2);
    const float hz = expf(-t3 * t3);
    const float cf = sigf(t4);

    const size_t pidx = (size_t)b * NPRED + i;
    boxesW[pidx * 4 + 0] = xs;
    boxesW[pidx * 4 + 1] = ys;
    boxesW[pidx * 4 + 2] = wz;
    boxesW[pidx * 4 + 3] = hz;
    confW[pidx] = cf;

    const float px1 = xs - wz * 0.5f, px2 = xs + wz * 0.5f;
    const float py1 = ys - hz * 0.5f, py2 = ys + hz * 0.5f;
    const float pa  = (px2 - px1) * (py2 - py1);

    float* __restrict__ grow = gous + pidx * GTB;
    float* __restrict__ irow = ious + pidx * GTB;
#pragma unroll
    for (int j = 0; j < GTB; ++j) {
        float iw = fminf(px2, gx2[j]) - fmaxf(px1, gx1[j]);
        float ih = fminf(py2, gy2[j]) - fmaxf(py1, gy1[j]);
        iw = fmaxf(iw, 0.0f);
        ih = fmaxf(ih, 0.0f);
        const float inter  = iw * ih;
        const float uni    = pa + gar[j] - inter;
        const float iou    = inter / (uni + EPSV);
        const float cw     = fmaxf(px2, gx2[j]) - fminf(px1, gx1[j]);
        const float ch     = fmaxf(py2, gy2[j]) - fminf(py1, gy1[j]);
        const float ac     = cw * ch;
        const float gou    = iou - (ac - uni) / (ac + EPSV);
        irow[j] = iou;
        grow[j] = gou;
    }
}

// ------------------------------------------------------------------
// Kernel 2: greedy matching (16 iterations), one block per batch.
// 16 sub-threads per GT column scan preds with first-max tie-break
// (== jnp.argmax). Thread 0 resolves winners (scatter-max semantics:
// larger key wins; ties -> larger j).  pdgt = matched local j, or -1.
// ------------------------------------------------------------------
__global__ void __launch_bounds__(256)
k_match(const float* __restrict__ gous, int* __restrict__ pdgt)
{
    const int b = blockIdx.x;
    const int t = threadIdx.x;

    __shared__ unsigned char maskS[NPRED];   // pred already matched
    __shared__ float rv[256];
    __shared__ int   ri[256];
    __shared__ float colv[GTB];
    __shared__ int   coli[GTB];
    __shared__ int   selectedS;              // bitmask of selected GT columns

    for (int i = t; i < NPRED; i += 256) {
        maskS[i] = 0;
        pdgt[(size_t)b * NPRED + i] = -1;
    }
    if (t == 0) selectedS = 0;
    __syncthreads();

    const int col = t & 15;
    const int sub = t >> 4;
    const float* __restrict__ G = gous + (size_t)b * NPRED * GTB;

    for (int iter = 0; iter < GTB; ++iter) {
        // --- per-column argmax over unmasked preds (strided by 16) ---
        float bv = NEGV;
        int   bi = -1;
        if (!((selectedS >> col) & 1)) {
            for (int i = sub; i < NPRED; i += 16) {
                if (i + 2048 < NPRED)
                    __builtin_prefetch(&G[(size_t)(i + 2048) * GTB + col], 0, 0);
                if (maskS[i]) continue;
                const float v = G[(size_t)i * GTB + col];
                if (v > bv) { bv = v; bi = i; }     // strict '>' keeps first max
            }
        }
        rv[t] = bv;
        ri[t] = bi;
        __syncthreads();

        // --- reduce the 16 sub-thread candidates per column ---
        if (t < GTB) {
            float v = NEGV;
            int  vi = -1;
            for (int s = 0; s < 16; ++s) {
                const float cv = rv[t + s * 16];
                const int   ci = ri[t + s * 16];
                if (ci < 0) continue;
                if (vi < 0 || cv > v || (cv == v && ci < vi)) { v = cv; vi = ci; }
            }
            colv[t] = v;
            coli[t] = vi;
        }
        __syncthreads();

        // --- winner resolution (matches reference scatter-max + besti) ---
        if (t == 0) {
            const int sel = selectedS;
            int winmask = 0;
            for (int j = 0; j < GTB; ++j) {
                if ((sel >> j) & 1) continue;
                if (coli[j] < 0) continue;           // inactive column
                bool win = true;
                for (int j2 = 0; j2 < GTB; ++j2) {
                    if (j2 == j || ((sel >> j2) & 1) || coli[j2] != coli[j] || coli[j2] < 0)
                        continue;
                    if (colv[j2] > colv[j] || (colv[j2] == colv[j] && j2 > j)) { win = false; break; }
                }
                if (win) winmask |= (1 << j);
            }
            for (int j = 0; j < GTB; ++j) {
                if ((winmask >> j) & 1) {
                    selectedS |= (1 << j);
                    maskS[coli[j]] = 1;
                    pdgt[(size_t)b * NPRED + coli[j]] = j;   // local GT index
                }
            }
        }
        __syncthreads();
    }
}

// ------------------------------------------------------------------
// Kernel 3: fallback argmax + per-pred loss terms + block reduction of
// 10 sums (padded to 16) into partial[block][16].  grid = NBLK3 (508);
// block 507 emits zeros so the partial matrix is 4-row aligned.
// ------------------------------------------------------------------
__global__ void __launch_bounds__(256)
k_parts(const float* __restrict__ gous, const float* __restrict__ ious,
        const float* __restrict__ boxesW, const float* __restrict__ confW,
        const int* __restrict__ pdgt, const float* __restrict__ gts,
        float* __restrict__ partial)
{
    const int t   = threadIdx.x;
    const int idx = blockIdx.x * 256 + t;

    float q[10];
#pragma unroll
    for (int k = 0; k < 10; ++k) q[k] = 0.0f;

    if (idx < NBATCH * NPRED) {
        const int b = idx / NPRED;
        const size_t row = (size_t)idx * GTB;

        const int jm = pdgt[idx];
        const float tp = (jm >= 0) ? 1.0f : 0.0f;
        int jloc = jm;
        if (jm < 0) {                       // fallback: argmax of gous row (first max)
            float bv = NEGV;
            jloc = 0;
#pragma unroll
            for (int j = 0; j < GTB; ++j) {
                const float v = gous[row + j];
                if (v > bv) { bv = v; jloc = j; }
            }
        }

        const float piou = ious[row + jloc];
        const float pgou = gous[row + jloc];
        const float* bx  = boxesW + (size_t)idx * 4;
        const float  cf  = confW[idx];
        const float* g   = gts + (size_t)(b * GTB + jloc) * 5;

        const float dx = bx[0] - g[1];
        const float dy = bx[1] - g[2];
        const float dw = bx[2] - g[3];
        const float dh = bx[3] - g[4];

        q[0] = tp * dx * dx;
        q[1] = tp * dy * dy;
        q[2] = tp * dw * dw;
        q[3] = tp * dh * dh;
        q[4] = tp * piou;
        q[5] = tp * pgou;
        q[6] = tp;                           // cnt
        const bool thr = piou > 0.5f;
        const float w  = (!thr || jm >= 0) ? 1.0f : 0.0f;   // ignore mask
        q[7] = w * cf * tp;                  // inter
        q[8] = w * cf * cf;                  // pd_area
        q[9] = w * tp;                       // gt_area
    }

    __shared__ float red[256];
    for (int k = 0; k < 10; ++k) {
        red[t] = q[k];
        __syncthreads();
        for (int s = 128; s > 0; s >>= 1) {
            if (t < s) red[t] += red[t + s];
            __syncthreads();
        }
        if (t == 0) partial[(size_t)blockIdx.x * 16 + k] = red[0];
        __syncthreads();
    }
    if (t < 6) partial[(size_t)blockIdx.x * 16 + 10 + t] = 0.0f;
}

// ------------------------------------------------------------------
// Kernel 4: single-wave WMMA reduction.  D = Ones(16x4) x B(4x16) + C
// gives exact f32 column sums of 4 partial rows per v_wmma; with A all
// ones the K-slot<->row mapping is permutation invariant, only the
// lane->column (N = lane & 15) mapping matters.  Lanes 0..15, VGPR0 of
// the accumulator hold D[0][n] = total sum of quantity n.
// ------------------------------------------------------------------
__global__ void __launch_bounds__(32)
k_final(const float* __restrict__ partial, float* __restrict__ outp)
{
    const int lane = threadIdx.x;
    const int n    = lane & 15;
    const int half = lane >> 4;

    const v2f aone = {1.0f, 1.0f};            // A = ones(16x4)
    v8f c = {0.f, 0.f, 0.f, 0.f, 0.f, 0.f, 0.f, 0.f};

    for (int chunk = 0; chunk < NBLK3 / 4; ++chunk) {   // 127 chunks of 4 rows
        const int r = chunk * 4 + half * 2;
        v2f bv;
        bv.x = partial[(size_t)(r + 0) * 16 + n];
        bv.y = partial[(size_t)(r + 1) * 16 + n];
        // D = A*B + C   (f32, 16x16x4)
        c = __builtin_amdgcn_wmma_f32_16x16x4_f32(
                /*neg_a=*/false, aone, /*neg_b=*/false, bv,
                /*c_mod=*/(short)0, c, /*reuse_a=*/false, /*reuse_b=*/false);
    }

    __shared__ float sums[16];
    if (lane < 16) sums[lane] = c[0];          // D[0][n]
    __syncthreads();

    if (lane == 0) {
        const float S0 = sums[0], S1 = sums[1], S2 = sums[2], S3 = sums[3];
        const float S4 = sums[4], S5 = sums[5], S6 = sums[6];
        const float S7 = sums[7], S8 = sums[8], S9 = sums[9];
        const float cnt = fmaxf(S6, 1.0f);
        const float lxy  = S0 / cnt + S1 / cnt;
        const float lwh  = S2 / cnt + S3 / cnt;
        const float liou = 1.0f - S4 / cnt;
        const float lgou = 1.0f - S5 / cnt;
        const float lconf = 1.0f - (2.0f * S7 + 1.0f) / (S8 + S9 + 1.0f);
        outp[0] = lxy;
        outp[1] = lwh;
        outp[2] = lconf;
        outp[3] = liou;
        outp[4] = lgou;
        outp[5] = lxy + lwh + lconf + lgou + liou;
    }
}

// ------------------------------------------------------------------
extern "C" void kernel_launch(void* const* d_in, const int* in_sizes, int n_in,
                              void* d_out, int out_size, void* d_ws, size_t ws_size,
                              hipStream_t stream)
{
    (void)in_sizes; (void)n_in; (void)out_size; (void)ws_size;

    const float* outT = (const float*)d_in[0];   // (16, 15, 52, 52)
    const float* gts  = (const float*)d_in[1];   // (256, 5)
    float* res = (float*)d_out;                  // 6 scalars

    float* ws = (float*)d_ws;
    size_t off = 0;
    float* gous   = ws + off; off += (size_t)NBATCH * NPRED * GTB;   // 2,076,672 f
    float* ious   = ws + off; off += (size_t)NBATCH * NPRED * GTB;   // 2,076,672 f
    float* boxesW = ws + off; off += (size_t)NBATCH * NPRED * 4;     //   519,168 f
    float* confW  = ws + off; off += (size_t)NBATCH * NPRED;         //   129,792 f
    int*   pdgt   = (int*)(ws + off); off += (size_t)NBATCH * NPRED; //   129,792 i
    float* partial = ws + off;                                       // 508*16 f

    k_decode<<<NBATCH * 32, 256, 0, stream>>>(outT, gts, gous, ious, boxesW, confW);
    k_match <<<NBATCH,      256, 0, stream>>>(gous, pdgt);
    k_parts <<<NBLK3,       256, 0, stream>>>(gous, ious, boxesW, confW, pdgt, gts, partial);
    k_final <<<1,            32, 0, stream>>>(partial, res);
}